// Gated_Conv_SNU_84421877170222
// MI455X (gfx1250) — compile-verified
//
#include <hip/hip_runtime.h>
#include <math.h>

// f32 WMMA fragments (wave32): A 16x4 f32 = 2 VGPRs, B 4x16 = 2 VGPRs, C/D 16x16 = 8 VGPRs
typedef float v2f __attribute__((ext_vector_type(2)));
typedef float v4f __attribute__((ext_vector_type(4)));
typedef float v8f __attribute__((ext_vector_type(8)));

#define BATCH   16
#define IMG_H   1024
#define IMG_W   1024
#define STRIP_C 256          // block output strip width (8 waves x 2 tiles x 16)
#define PATCH_R 20           // 16 + 4 halo rows
#define PATCH_C (STRIP_C+4)  // 260: strip + 4 halo cols
#define NFRAG   4800         // 6 convs * 25 chunks * 32 lanes

// ---------------------------------------------------------------------------
// Setup kernel: build all banded-B WMMA fragments once, in B-register layout.
// Fragment (q, r, c, lane): B[t, j] = w_q[r, t-j] for t in the K=4 chunk c.
// VGPR0 holds K = 4c + 2*kh, VGPR1 holds K = 4c + 2*kh + 1  (kh = lane>>4).
// ---------------------------------------------------------------------------
__global__ __launch_bounds__(256)
void build_bfrag_kernel(const float* __restrict__ Wi, const float* __restrict__ Wf,
                        const float* __restrict__ Wx, const float* __restrict__ Ri,
                        const float* __restrict__ Rf, const float* __restrict__ Wy,
                        v2f* __restrict__ ws)
{
    const int idx = blockIdx.x * 256 + threadIdx.x;
    if (idx >= NFRAG) return;
    const int lane = idx & 31;
    const int t    = idx >> 5;      // q*25 + r*5 + c
    const int c    = t % 5;
    const int rq   = t / 5;
    const int r    = rq % 5;
    const int q    = rq / 5;        // 0:Wi 1:Wf 2:Wx 3:Ri 4:Rf 5:Wy

    const float* w = Wi;
    if (q == 1) w = Wf;
    if (q == 2) w = Wx;
    if (q == 3) w = Ri;
    if (q == 4) w = Rf;
    if (q == 5) w = Wy;

    const int j  = lane & 15;
    const int kh = lane >> 4;
    const int t0 = 4 * c + 2 * kh;
    const int d0 = t0 - j;
    const int d1 = d0 + 1;

    v2f f;
    f.x = (d0 >= 0 && d0 < 5) ? w[5 * r + d0] : 0.0f;
    f.y = (d1 >= 0 && d1 < 5) ? w[5 * r + d1] : 0.0f;
    ws[idx] = f;
}

// ---------------------------------------------------------------------------
// Three convs sharing one source patch, two 16x16 tiles per wave.
// Per (r,c) chunk: 2 A-loads + 3 B-loads feed 6 WMMAs.
// ---------------------------------------------------------------------------
__device__ __forceinline__ void conv_group3x2(
    const float (*src)[PATCH_C], const v2f* __restrict__ ldsB,
    int qbase, int cb0, int lane,
    v8f& aI0, v8f& aI1, v8f& aF0, v8f& aF1, v8f& aT0, v8f& aT1)
{
    const int M   = lane & 15;
    const int kh  = lane >> 4;
    const int cb1 = cb0 + 16;
#pragma unroll
    for (int r = 0; r < 5; ++r) {
#pragma unroll
        for (int c = 0; c < 5; ++c) {
            const int t0 = 4 * c + 2 * kh;
            v2f A0, A1;
            A0.x = src[M + r][cb0 + t0];
            A0.y = src[M + r][cb0 + t0 + 1];
            A1.x = src[M + r][cb1 + t0];
            A1.y = src[M + r][cb1 + t0 + 1];

            const int fi = (qbase * 25 + r * 5 + c) * 32 + lane;
            const v2f B0 = ldsB[fi];          // gate-i weights (Wi / Ri)
            const v2f B1 = ldsB[fi + 800];    // gate-f weights (Wf / Rf)
            const v2f B2 = ldsB[fi + 1600];   // membrane weights (Wx / Wy)

            aI0 = __builtin_amdgcn_wmma_f32_16x16x4_f32(false, A0, false, B0, (short)0, aI0, false, false);
            aI1 = __builtin_amdgcn_wmma_f32_16x16x4_f32(false, A1, false, B0, (short)0, aI1, false, false);
            aF0 = __builtin_amdgcn_wmma_f32_16x16x4_f32(false, A0, false, B1, (short)0, aF0, false, false);
            aF1 = __builtin_amdgcn_wmma_f32_16x16x4_f32(false, A1, false, B1, (short)0, aF1, false, false);
            aT0 = __builtin_amdgcn_wmma_f32_16x16x4_f32(false, A0, false, B2, (short)0, aT0, false, false);
            aT1 = __builtin_amdgcn_wmma_f32_16x16x4_f32(false, A1, false, B2, (short)0, aT1, false, false);
        }
    }
}

__global__ __launch_bounds__(256)
void snu_wmma_kernel(
    const float* __restrict__ x, const float* __restrict__ s, const float* __restrict__ y,
    const float* __restrict__ Wi_b, const float* __restrict__ Wf_b,
    const float* __restrict__ Wx_b, const float* __restrict__ Wy_b,
    const float* __restrict__ Ri_b, const float* __restrict__ Rf_b,
    const float* __restrict__ bscal, const v2f* __restrict__ wsB,
    float* __restrict__ out)
{
    __shared__ float ldsX[PATCH_R][PATCH_C];
    __shared__ float ldsY[PATCH_R][PATCH_C];
    __shared__ v2f   ldsB[NFRAG];

    const int tid  = threadIdx.x;
    const int wave = tid >> 5;
    const int lane = tid & 31;

    const int bx = blockIdx.x;     // 256-col strip index
    const int by = blockIdx.y;     // 16-row tile index
    const int bz = blockIdx.z;     // batch

    const long imgOff = (long)bz * IMG_H * IMG_W;
    const int gy0 = by * 16 - 2;
    const int gx0 = bx * STRIP_C - 2;

    // Early prefetch of the s block this wave consumes in the epilogue.
    {
        const long po = imgOff + (long)(by * 16) * IMG_W + bx * STRIP_C + wave * 32;
        __builtin_prefetch(&s[po], 0, 2);
    }

    // Copy precomputed B fragments (L2-hot, 38.4 KB) into LDS as float4.
    for (int idx = tid; idx < NFRAG / 2; idx += 256) {
        ((v4f*)ldsB)[idx] = ((const v4f*)wsB)[idx];
    }

    // Cooperative zero-padded fill of x/y halo patches.
    for (int idx = tid; idx < PATCH_R * PATCH_C; idx += 256) {
        const int pr = idx / PATCH_C;
        const int pc = idx - pr * PATCH_C;
        const int gr = gy0 + pr;
        const int gc = gx0 + pc;
        float vx = 0.0f, vy = 0.0f;
        if (gr >= 0 && gr < IMG_H && gc >= 0 && gc < IMG_W) {
            const long o = imgOff + (long)gr * IMG_W + gc;
            vx = x[o];
            vy = y[o];
        }
        ldsX[pr][pc] = vx;
        ldsY[pr][pc] = vy;
    }
    __syncthreads();

    const int cb0 = wave * 32;     // patch col base of this wave's first tile

    v8f acc_i0 = {0,0,0,0,0,0,0,0}, acc_i1 = {0,0,0,0,0,0,0,0};
    v8f acc_f0 = {0,0,0,0,0,0,0,0}, acc_f1 = {0,0,0,0,0,0,0,0};
    v8f acc_x0 = {0,0,0,0,0,0,0,0}, acc_x1 = {0,0,0,0,0,0,0,0};
    v8f acc_w0 = {0,0,0,0,0,0,0,0}, acc_w1 = {0,0,0,0,0,0,0,0};

    conv_group3x2(ldsX, ldsB, 0, cb0, lane, acc_i0, acc_i1, acc_f0, acc_f1, acc_x0, acc_x1);
    conv_group3x2(ldsY, ldsB, 3, cb0, lane, acc_i0, acc_i1, acc_f0, acc_f1, acc_w0, acc_w1);

    // Epilogue: sigmoid gates, ELU membrane, Heaviside spike.
    const float wib = Wi_b[0] + Ri_b[0];
    const float wfb = Wf_b[0] + Rf_b[0];
    const float wxb = Wx_b[0];
    const float wyb = Wy_b[0];
    const float bb  = bscal[0];

    const int jj = lane & 15;
    const int kh = lane >> 4;

#pragma unroll
    for (int t = 0; t < 2; ++t) {
        const v8f& ai = t ? acc_i1 : acc_i0;
        const v8f& af = t ? acc_f1 : acc_f0;
        const v8f& ax = t ? acc_x1 : acc_x0;
        const v8f& aw = t ? acc_w1 : acc_w0;
        const int colG = bx * STRIP_C + wave * 32 + t * 16 + jj;
        const int colP = wave * 32 + t * 16 + jj + 2;   // patch col of output pixel
#pragma unroll
        for (int v = 0; v < 8; ++v) {
            const int irow = v + 8 * kh;                 // tile row (C/D layout)
            const long o   = imgOff + (long)(by * 16 + irow) * IMG_W + colG;

            const float gi = 1.0f / (1.0f + __expf(-(ai[v] + wib)));
            const float gf = 1.0f / (1.0f + __expf(-(af[v] + wfb)));

            const float sv = s[o];
            const float yv = ldsY[irow + 2][colP];

            const float pre = ax[v] + wxb + gi * (aw[v] + wyb) + gf * sv * (1.0f - yv);
            const float sn  = (pre > 0.0f) ? pre : (__expf(pre) - 1.0f);
            out[o] = ((sn + bb) > 0.0f) ? 1.0f : 0.0f;
        }
    }
}

extern "C" void kernel_launch(void* const* d_in, const int* in_sizes, int n_in,
                              void* d_out, int out_size, void* d_ws, size_t ws_size,
                              hipStream_t stream) {
    (void)in_sizes; (void)n_in; (void)out_size; (void)ws_size;
    // Inputs: 0:x 1:s 2:y 3:Wx_w 4:Wx_b 5:Wi_w 6:Wi_b 7:Wf_w 8:Wf_b
    //         9:Wy_w 10:Wy_b 11:Ri_w 12:Ri_b 13:Rf_w 14:Rf_b 15:b
    v2f* wsB = (v2f*)d_ws;

    build_bfrag_kernel<<<(NFRAG + 255) / 256, 256, 0, stream>>>(
        (const float*)d_in[5],  (const float*)d_in[7],  (const float*)d_in[3],
        (const float*)d_in[11], (const float*)d_in[13], (const float*)d_in[9],
        wsB);

    dim3 grid(IMG_W / STRIP_C, IMG_H / 16, BATCH);   // 4 x 64 x 16 blocks
    dim3 block(256);                                 // 8 wave32 per block
    snu_wmma_kernel<<<grid, block, 0, stream>>>(
        (const float*)d_in[0],  (const float*)d_in[1],  (const float*)d_in[2],
        (const float*)d_in[6],  (const float*)d_in[8],  (const float*)d_in[4],
        (const float*)d_in[10], (const float*)d_in[12], (const float*)d_in[14],
        (const float*)d_in[15], wsB, (float*)d_out);
}